// ParallelCheby2D_29540785062618
// MI455X (gfx1250) — compile-verified
//
#include <hip/hip_runtime.h>

// ---------------------------------------------------------------------------
// ParallelCheby2D on MI455X (gfx1250), reformulated as a 32x32 fp32 WMMA GEMM.
//
//   V[8*d+p, t] = T_p(|x[t-d]|)          d=0..3, p=0..7   (32 features)
//   Wbig[8*d0+p, 8*d2+q] = w[4*d0+d2, p, q]               (32x32)
//   Y = Wbig * V   (real & imag)         <- V_WMMA_F32_16X16X4_F32
//   s[d0,t] = sum_p V[8*d0+p,t] * Y[8*d0+p,t]
//   out[t]  = sum_{d0} x[t-d0] * (s_r[d0,t] + i*s_i[d0,t])
// ---------------------------------------------------------------------------

typedef __attribute__((ext_vector_type(2))) float v2f;
typedef __attribute__((ext_vector_type(8))) float v8f;

#define ORDER 8
#define NT    512            // time samples per workgroup
#define PAD   3              // max delay
#define NLOC  (NT + PAD)     // staged samples (t0-3 .. t0+NT-1)

__global__ __launch_bounds__(256)
void cheb2d_wmma_kernel(const float* __restrict__ xr,
                        const float* __restrict__ xi,
                        const float* __restrict__ wr,
                        const float* __restrict__ wi,
                        float* __restrict__ out,
                        int T)
{
    __shared__ float sC[ORDER][NLOC];   // Chebyshev features C[p][i], i -> t0-PAD+i
    __shared__ float sXr[NLOC];
    __shared__ float sXi[NLOC];

    const int tid = threadIdx.x;
    const int t0  = blockIdx.x * NT;
    const int b   = blockIdx.y;
    const float* xr_b = xr + (size_t)b * T;
    const float* xi_b = xi + (size_t)b * T;

    // ---- Phase 1: stage x, |x|, Chebyshev features T_0..T_7 into LDS -------
    for (int i = tid; i < NLOC; i += 256) {
        int g = t0 - PAD + i;
        float re = 0.f, im = 0.f;
        if (g >= 0 && g < T) { re = xr_b[g]; im = xi_b[g]; }
        sXr[i] = re;
        sXi[i] = im;
        float a   = __builtin_sqrtf(re * re + im * im);
        float tm2 = 1.0f, tm1 = a;
        sC[0][i] = tm2;
        sC[1][i] = tm1;
        #pragma unroll
        for (int p = 2; p < ORDER; ++p) {
            float tk = 2.0f * a * tm1 - tm2;
            sC[p][i] = tk;
            tm2 = tm1;
            tm1 = tk;
        }
    }
    // Speculative prefetch of the next tile (emits global_prefetch_b8).
    {
        int gn = t0 + NT + tid;
        if (gn < T) {
            __builtin_prefetch(&xr_b[gn], 0, 0);
            __builtin_prefetch(&xi_b[gn], 0, 0);
        }
    }
    __syncthreads();

    const int lane = tid & 31;
    const int wave = tid >> 5;
    const int half = lane >> 4;     // 0: lanes 0-15, 1: lanes 16-31
    const int l15  = lane & 15;

    // ---- Phase 2: load A-fragments (Wbig, 32x32, re+im) into registers -----
    // A 16x4 f32 layout: lane l<16 -> (M=l, K=0..1), lane l+16 -> (M=l, K=2..3)
    v2f Ar[2][8], Ai[2][8];
    #pragma unroll
    for (int mt = 0; mt < 2; ++mt) {
        const int row = mt * 16 + l15;          // Wbig row = 8*d0 + p
        const int d0  = row >> 3;
        const int p   = row & 7;
        #pragma unroll
        for (int k = 0; k < 8; ++k) {
            const int c0 = 4 * k + 2 * half;    // Wbig col = 8*d2 + q
            const int c1 = c0 + 1;
            const int i0 = (4 * d0 + (c0 >> 3)) * 64 + p * 8 + (c0 & 7);
            const int i1 = (4 * d0 + (c1 >> 3)) * 64 + p * 8 + (c1 & 7);
            v2f ar; ar.x = wr[i0]; ar.y = wr[i1];
            v2f ai; ai.x = wi[i0]; ai.y = wi[i1];
            Ar[mt][k] = ar;
            Ai[mt][k] = ai;
        }
    }

    // ---- Phase 3: per 16-column tile -> 32 WMMAs, contract, combine --------
    // 8 waves x (NT/16/8 = 4) tiles each.
    #pragma unroll
    for (int it = 0; it < NT / (16 * 8); ++it) {
        const int j = (wave + 8 * it) * 16;     // column offset in this block

        v8f acc_r0 = {}, acc_r1 = {}, acc_i0 = {}, acc_i1 = {};

        #pragma unroll
        for (int k = 0; k < 8; ++k) {
            // B 4x16 f32: lane l<16 supplies K rows {4k,4k+1}, lane l+16 {4k+2,4k+3}
            const int r0 = 4 * k + 2 * half;
            const int r1 = r0 + 1;
            v2f Bf;
            Bf.x = sC[r0 & 7][j + l15 + PAD - (r0 >> 3)];
            Bf.y = sC[r1 & 7][j + l15 + PAD - (r1 >> 3)];

            acc_r0 = __builtin_amdgcn_wmma_f32_16x16x4_f32(
                false, Ar[0][k], false, Bf, (short)0, acc_r0, false, false);
            acc_r1 = __builtin_amdgcn_wmma_f32_16x16x4_f32(
                false, Ar[1][k], false, Bf, (short)0, acc_r1, false, false);
            acc_i0 = __builtin_amdgcn_wmma_f32_16x16x4_f32(
                false, Ai[0][k], false, Bf, (short)0, acc_i0, false, false);
            acc_i1 = __builtin_amdgcn_wmma_f32_16x16x4_f32(
                false, Ai[1][k], false, Bf, (short)0, acc_i1, false, false);
        }

        // D 16x16 f32: lane l<16 holds M=v (rows 0-7), lane l+16 holds M=8+v.
        //   acc*_0 -> delay block d0 = half ; acc*_1 -> d0 = 2 + half.
        float pr = 0.f, pi = 0.f;
        #pragma unroll
        for (int mt = 0; mt < 2; ++mt) {
            const int d0 = 2 * mt + half;
            const int li = j + l15 + PAD - d0;
            float sr = 0.f, si = 0.f;
            #pragma unroll
            for (int v = 0; v < 8; ++v) {
                const float vv = sC[v][li];     // V[8*d0+v, t] = C[v][t-d0]
                sr += vv * (mt == 0 ? acc_r0[v] : acc_r1[v]);
                si += vv * (mt == 0 ? acc_i0[v] : acc_i1[v]);
            }
            const float xre = sXr[li], xim = sXi[li];
            pr += xre * sr - xim * si;          // x[t-d0] * g[d0,t] (complex)
            pi += xre * si + xim * sr;
        }

        // Merge the two half-wave partials for the same column (lanes l, l+16).
        pr += __shfl_xor(pr, 16, 32);
        pi += __shfl_xor(pi, 16, 32);

        if (half == 0) {
            const int t = t0 + j + l15;
            if (t < T) {
                const size_t o = 2 * ((size_t)b * T + t);
                out[o + 0] = pr;   // complex64 interleaved (re, im)
                out[o + 1] = pi;
            }
        }
    }
}

extern "C" void kernel_launch(void* const* d_in, const int* in_sizes, int n_in,
                              void* d_out, int out_size, void* d_ws, size_t ws_size,
                              hipStream_t stream) {
    (void)n_in; (void)out_size; (void)d_ws; (void)ws_size;
    const float* xr = (const float*)d_in[0];   // [B,1,T] fp32
    const float* xi = (const float*)d_in[1];   // [B,1,T] fp32
    const float* wr = (const float*)d_in[2];   // [16,8,8] fp32
    const float* wi = (const float*)d_in[3];   // [16,8,8] fp32
    float* out = (float*)d_out;                // [B,T] complex64 -> interleaved f32

    const int B = 4;
    const int T = in_sizes[0] / B;

    dim3 grid((T + NT - 1) / NT, B);
    dim3 block(256);
    cheb2d_wmma_kernel<<<grid, block, 0, stream>>>(xr, xi, wr, wi, out, T);
}